// PtModule_76166950027878
// MI455X (gfx1250) — compile-verified
//
#include <hip/hip_runtime.h>

// Problem shape from the reference: x is [8192, 4096] float32.
#define COLS   4096
#define TPB    256                    // 8 waves of 32 (CDNA5 is wave32)
#define WAVES  (TPB / 32)
#define CHUNKS 4                      // 4 x float4 per thread: 4*4*256 = 4096 floats

// ---- CDNA5 async global->LDS path (ASYNCcnt-tracked), probe-guarded ----
#if __has_builtin(__builtin_amdgcn_global_load_async_to_lds_b128)
#define HAVE_ASYNC_LDS 1
#else
#define HAVE_ASYNC_LDS 0
#endif

// Exact pointee type from the round-1 diagnostic:
//   parameter 0: '__attribute__((__vector_size__(4 * sizeof(int)))) int __device__ *'
typedef int v4i __attribute__((vector_size(16)));
typedef __attribute__((address_space(1))) v4i* gv4i_ptr;   // global int4*
typedef __attribute__((address_space(3))) v4i* lv4i_ptr;   // LDS int4*

__global__ __launch_bounds__(TPB)
void row_scale_cond(const float* __restrict__ x, float* __restrict__ y, int rows) {
    __shared__ float srow[COLS];      // 16 KB row staging (WGP has 320 KB LDS)
    __shared__ float swsum[WAVES];

    const int tid  = threadIdx.x;
    const int lane = tid & 31;
    const int wid  = tid >> 5;
    const size_t row = blockIdx.x;
    if ((int)row >= rows) return;

    const float* __restrict__ xr = x + row * (size_t)COLS;
    float*       __restrict__ yr = y + row * (size_t)COLS;

    // ---- Stage the full row into LDS (one DRAM read total) ----
#if HAVE_ASYNC_LDS
    #pragma unroll
    for (int c = 0; c < CHUNKS; ++c) {
        const int idx = c * (TPB * 4) + tid * 4;   // lane-contiguous b128 per issue
        __builtin_amdgcn_global_load_async_to_lds_b128(
            (gv4i_ptr)(xr + idx), (lv4i_ptr)(srow + idx), /*offset=*/0, /*cpol=*/0);
    }
    // New split dependency counter on gfx1250: async LDS DMAs use ASYNCcnt.
    #if __has_builtin(__builtin_amdgcn_s_wait_asynccnt)
    __builtin_amdgcn_s_wait_asynccnt(0);
    #else
    asm volatile("s_wait_asynccnt 0" ::: "memory");
    #endif
#else
    #pragma unroll
    for (int c = 0; c < CHUNKS; ++c) {
        const int idx = c * (TPB * 4) + tid * 4;
        *(float4*)(srow + idx) = *(const float4*)(xr + idx);
    }
#endif
    __syncthreads();

    // ---- Per-thread partial sums; keep data in registers for the store phase ----
    float4 v[CHUNKS];
    float psum = 0.f;
    #pragma unroll
    for (int c = 0; c < CHUNKS; ++c) {
        v[c] = *(const float4*)(srow + c * (TPB * 4) + tid * 4);  // ds_load_b128
        psum += (v[c].x + v[c].y) + (v[c].z + v[c].w);
    }

    // ---- wave32 xor-shuffle reduction (NOT 64 — CDNA5 is wave32) ----
    #pragma unroll
    for (int off = 16; off > 0; off >>= 1)
        psum += __shfl_xor(psum, off, 32);

    if (lane == 0) swsum[wid] = psum;
    __syncthreads();

    float total = 0.f;
    #pragma unroll
    for (int w = 0; w < WAVES; ++w) total += swsum[w];

    // row_sum(y) = 2 * sum(x); subtract 1 everywhere if it exceeds 10
    const float sub = (2.0f * total > 10.0f) ? 1.0f : 0.0f;

    // ---- Rewrite from registers: no second DRAM read ----
    #pragma unroll
    for (int c = 0; c < CHUNKS; ++c) {
        float4 o;
        o.x = 2.0f * v[c].x - sub;
        o.y = 2.0f * v[c].y - sub;
        o.z = 2.0f * v[c].z - sub;
        o.w = 2.0f * v[c].w - sub;
        *(float4*)(yr + c * (TPB * 4) + tid * 4) = o;   // global_store_b128
    }
}

extern "C" void kernel_launch(void* const* d_in, const int* in_sizes, int n_in,
                              void* d_out, int out_size, void* d_ws, size_t ws_size,
                              hipStream_t stream) {
    (void)n_in; (void)out_size; (void)d_ws; (void)ws_size;
    const float* x = (const float*)d_in[0];
    float* y = (float*)d_out;
    const int rows = in_sizes[0] / COLS;   // 8192 for the reference shapes
    row_scale_cond<<<dim3(rows), dim3(TPB), 0, stream>>>(x, y, rows);
}